// RBF_22204980920839
// MI455X (gfx1250) — compile-verified
//
#include <hip/hip_runtime.h>

typedef __attribute__((ext_vector_type(2))) float v2f;
typedef __attribute__((ext_vector_type(4))) float f32x4;
typedef __attribute__((ext_vector_type(8))) float v8f;

constexpr int kN = 2048;
constexpr int kM = 2048;
constexpr int kD = 8;

__device__ __forceinline__ void st_nt4(float* p, float a, float b, float c, float d) {
    f32x4 v;
    v[0] = a; v[1] = b; v[2] = c; v[3] = d;
    __builtin_nontemporal_store(v, reinterpret_cast<f32x4*>(p));
}

// One wave per 16x16 (i,j) tile. WMMA f32 16x16x4 (x2, K=8) computes the
// Gram cross-term; VALU finishes exp + outer products; NT b128 stores stream
// the 1.36 GB of output without polluting L2.
__global__ __launch_bounds__(32) void rbf_wmma_kernel(
    const float* __restrict__ x,
    const float* __restrict__ xx,
    const float* __restrict__ scale_free,
    const float* __restrict__ variance_free,
    float* __restrict__ out)
{
    __shared__ float xsh[16][kD];   // x tile rows, pre-scaled by iscale
    __shared__ float xxsh[16][kD];  // xx tile rows, pre-scaled by iscale

    const int lane = threadIdx.x;   // 0..31
    const int half = lane >> 4;     // 0 or 1
    const int m    = lane & 15;     // 0..15
    const int i0   = blockIdx.y * 16;
    const int j0   = blockIdx.x * 16;

    // Per-lane hyperparameters (9 floats, L2/L1 resident).
    float iscale[kD], is2[kD];
#pragma unroll
    for (int d = 0; d < kD; ++d) {
        iscale[d] = __expf(-scale_free[d]);
        is2[d]    = iscale[d] * iscale[d];
    }
    const float variance = __expf(variance_free[0]);

    // Cooperative tile load: each lane loads one float4 of x and of xx.
    {
        const int r = lane >> 1;        // 0..15
        const int c = (lane & 1) * 4;   // 0 or 4
        const float4 xv = *reinterpret_cast<const float4*>(x + (size_t)(i0 + r) * kD + c);
        xsh[r][c + 0] = xv.x * iscale[c + 0];
        xsh[r][c + 1] = xv.y * iscale[c + 1];
        xsh[r][c + 2] = xv.z * iscale[c + 2];
        xsh[r][c + 3] = xv.w * iscale[c + 3];
        const float4 xxv = *reinterpret_cast<const float4*>(xx + (size_t)(j0 + r) * kD + c);
        xxsh[r][c + 0] = xxv.x * iscale[c + 0];
        xxsh[r][c + 1] = xxv.y * iscale[c + 1];
        xxsh[r][c + 2] = xxv.z * iscale[c + 2];
        xxsh[r][c + 3] = xxv.w * iscale[c + 3];
    }
    __syncthreads();

    // A: 16x4 f32 layout — lanes 0-15 hold (M=lane, K=0/1), lanes 16-31 (M=lane-16, K=2/3).
    // B: 4x16 f32 layout — mirrored (N=lane, same K split). Two WMMAs cover K=0..7.
    const int k0 = 2 * half;
    v2f a_lo, a_hi, b_lo, b_hi;
    a_lo[0] = xsh[m][k0];      a_lo[1] = xsh[m][k0 + 1];
    a_hi[0] = xsh[m][k0 + 4];  a_hi[1] = xsh[m][k0 + 5];
    b_lo[0] = xxsh[m][k0];     b_lo[1] = xxsh[m][k0 + 1];
    b_hi[0] = xxsh[m][k0 + 4]; b_hi[1] = xxsh[m][k0 + 5];

    v8f acc = {};
    acc = __builtin_amdgcn_wmma_f32_16x16x4_f32(false, a_lo, false, b_lo,
                                                (short)0, acc, false, false);
    acc = __builtin_amdgcn_wmma_f32_16x16x4_f32(false, a_hi, false, b_hi,
                                                (short)0, acc, false, false);

    // This lane's xx column (j = j0 + m) and its squared norm.
    float xxc[kD];
    float sxx = 0.f;
#pragma unroll
    for (int d = 0; d < kD; ++d) { xxc[d] = xxsh[m][d]; sxx += xxc[d] * xxc[d]; }

    float* const out_cov = out;                                  // [N,M]
    float* const out_dx  = out + (size_t)kN * kM;                // [N,M,8]
    float* const out_dxx = out_dx + (size_t)kN * kM * kD;        // [N,M,8]
    float* const out_dd  = out_dxx + (size_t)kN * kM * kD;       // [N,M,8,8]

    const int j = j0 + m;
#pragma unroll
    for (int v = 0; v < 8; ++v) {
        const int ir = half * 8 + v;          // C/D layout: VGPR v -> row half*8+v
        const int i  = i0 + ir;

        float xr[kD], sx = 0.f;
#pragma unroll
        for (int d = 0; d < kD; ++d) { xr[d] = xsh[ir][d]; sx += xr[d] * xr[d]; }

        // -0.5*||r||^2 = dot - 0.5*(||xs||^2 + ||xxs||^2)
        const float cov = __expf(acc[v] - 0.5f * (sx + sxx)) * variance;

        float rr[kD], cr[kD];
#pragma unroll
        for (int d = 0; d < kD; ++d) rr[d] = (xr[d] - xxc[d]) * iscale[d]; // r*iscale
#pragma unroll
        for (int d = 0; d < kD; ++d) cr[d] = cov * rr[d];

        const size_t pij = (size_t)i * kM + j;
        __builtin_nontemporal_store(cov, out_cov + pij);

        st_nt4(out_dx  + pij * 8,     -cr[0], -cr[1], -cr[2], -cr[3]);
        st_nt4(out_dx  + pij * 8 + 4, -cr[4], -cr[5], -cr[6], -cr[7]);
        st_nt4(out_dxx + pij * 8,      cr[0],  cr[1],  cr[2],  cr[3]);
        st_nt4(out_dxx + pij * 8 + 4,  cr[4],  cr[5],  cr[6],  cr[7]);

        float* dd = out_dd + pij * 64;  // 256 B contiguous per pair, lane-consecutive
#pragma unroll
        for (int a = 0; a < 8; ++a) {
            float w[8];
#pragma unroll
            for (int b = 0; b < 8; ++b) {
                float val = -rr[a] * rr[b];
                if (a == b) val += is2[a];
                w[b] = val * cov;
            }
            st_nt4(dd + a * 8,     w[0], w[1], w[2], w[3]);
            st_nt4(dd + a * 8 + 4, w[4], w[5], w[6], w[7]);
        }
    }
}

extern "C" void kernel_launch(void* const* d_in, const int* in_sizes, int n_in,
                              void* d_out, int out_size, void* d_ws, size_t ws_size,
                              hipStream_t stream) {
    const float* x             = (const float*)d_in[0];
    const float* xx            = (const float*)d_in[1];
    const float* scale_free    = (const float*)d_in[2];
    const float* variance_free = (const float*)d_in[3];
    float* out = (float*)d_out;

    dim3 grid(kM / 16, kN / 16);  // 128 x 128 tiles, one wave each
    rbf_wmma_kernel<<<grid, 32, 0, stream>>>(x, xx, scale_free, variance_free, out);
}